// Attention_25262997635378
// MI455X (gfx1250) — compile-verified
//
#include <hip/hip_runtime.h>
#include <stdint.h>

// Problem constants (from reference): N=262144, QDIM=256, KDIM=512
#define NKEYS 262144
#define QD 256
#define KD 512

#define NB_SCORE 2048   // blocks for score pass (8 waves each -> 16384 waves, 16 rows/wave)
#define NB_EXP   1024   // blocks for exp pass
#define NB_OUT   512    // blocks for value pass (512 rows each, 64 rows/wave)

typedef __attribute__((ext_vector_type(2))) float v2f;
typedef __attribute__((ext_vector_type(8))) float v8f;

// ---------------------------------------------------------------------------
// K1: q = W @ query + b via V_WMMA_F32_16X16X4_F32.
// One block, 8 waves; wave w handles 4 row-tiles of 16 rows (8*4*16 = 512).
// A tile = W[m0..m0+15][k..k+3]; B = query[k..k+3] broadcast over all 16
// columns, so every column of D holds the q-tile; lanes 0/16 extract M rows.
// ---------------------------------------------------------------------------
__global__ __launch_bounds__(256) void qproj_wmma(const float* __restrict__ W,
                                                  const float* __restrict__ query,
                                                  const float* __restrict__ bias,
                                                  float* __restrict__ q) {
  const int tid   = threadIdx.x;
  const int wave  = tid >> 5;
  const int lane  = tid & 31;
  const int mrow  = lane & 15;
  const int khalf = (lane >> 4) << 1;    // lanes 16-31 handle K=k+2,k+3

  for (int t = 0; t < 4; ++t) {
    const int m0 = (wave * 4 + t) * 16;
    const float* wrow = W + (size_t)(m0 + mrow) * QD;
    v8f c = {};
    for (int k = 0; k < QD; k += 4) {
      const int kk = k + khalf;
      v2f a, b;
      a.x = wrow[kk];
      a.y = wrow[kk + 1];
      b.x = query[kk];
      b.y = query[kk + 1];
      c = __builtin_amdgcn_wmma_f32_16x16x4_f32(
          false, a, false, b, (short)0, c, false, false);
    }
    if (lane == 0) {
#pragma unroll
      for (int r = 0; r < 8; ++r) q[m0 + r] = c[r] + bias[m0 + r];
    } else if (lane == 16) {
#pragma unroll
      for (int r = 0; r < 8; ++r) q[m0 + 8 + r] = c[r] + bias[m0 + 8 + r];
    }
  }
}

// ---------------------------------------------------------------------------
// K2: scores[i] = key[i,:] . q   (streams 512 MB of key)
// One row per wave per grid-stride iteration, b128 loads, q in registers,
// full butterfly reduce, per-block max. Prefetch one stride ahead into L2.
// ---------------------------------------------------------------------------
__global__ __launch_bounds__(256) void score_kernel(const float* __restrict__ key,
                                                    const float* __restrict__ q,
                                                    float* __restrict__ scores,
                                                    float* __restrict__ blockmax) {
  __shared__ float smax[8];
  const int lane  = threadIdx.x & 31;
  const int wid   = threadIdx.x >> 5;
  const int gwave = blockIdx.x * 8 + wid;
  const int nwave = gridDim.x * 8;

  float4 qr[4];
#pragma unroll
  for (int s = 0; s < 4; ++s)
    qr[s] = *(const float4*)&q[s * 128 + lane * 4];

  float wmax = -3.4e38f;
  for (int row = gwave; row < NKEYS; row += nwave) {
    const float* kr = key + (size_t)row * KD;
    __builtin_prefetch(kr + (size_t)nwave * KD + lane * 4, 0, 1);
    float sum = 0.f;
#pragma unroll
    for (int s = 0; s < 4; ++s) {
      const float4 kv = *(const float4*)&kr[s * 128 + lane * 4];
      sum += kv.x * qr[s].x + kv.y * qr[s].y + kv.z * qr[s].z + kv.w * qr[s].w;
    }
#pragma unroll
    for (int off = 16; off >= 1; off >>= 1)
      sum += __shfl_xor(sum, off, 32);
    if (lane == 0) scores[row] = sum;
    wmax = fmaxf(wmax, sum);
  }
  if (lane == 0) smax[wid] = wmax;
  __syncthreads();
  if (threadIdx.x == 0) {
    float m = smax[0];
#pragma unroll
    for (int i = 1; i < 8; ++i) m = fmaxf(m, smax[i]);
    blockmax[blockIdx.x] = m;
  }
}

// ---------------------------------------------------------------------------
// K3: each block re-reduces the 2048 block maxima (8 KB, L2-resident) to get
// the global max, then scores[i] <- exp(scores[i]-gmax) with per-block sums.
// ---------------------------------------------------------------------------
__global__ __launch_bounds__(256) void expsum_kernel(float* __restrict__ scores,
                                                     const float* __restrict__ blockmax,
                                                     float* __restrict__ blocksum) {
  __shared__ float s[256];
  // fused global-max reduction
  float m = -3.4e38f;
  for (int i = threadIdx.x; i < NB_SCORE; i += 256) m = fmaxf(m, blockmax[i]);
  s[threadIdx.x] = m;
  __syncthreads();
  for (int off = 128; off >= 1; off >>= 1) {
    if (threadIdx.x < off) s[threadIdx.x] = fmaxf(s[threadIdx.x], s[threadIdx.x + off]);
    __syncthreads();
  }
  const float gmax = s[0];
  __syncthreads();

  float acc = 0.f;
  for (int i = blockIdx.x * 256 + threadIdx.x; i < NKEYS; i += gridDim.x * 256) {
    const float w = __expf(scores[i] - gmax);
    scores[i] = w;
    acc += w;
  }
  s[threadIdx.x] = acc;
  __syncthreads();
  for (int off = 128; off >= 1; off >>= 1) {
    if (threadIdx.x < off) s[threadIdx.x] += s[threadIdx.x + off];
    __syncthreads();
  }
  if (threadIdx.x == 0) blocksum[blockIdx.x] = s[0];
}

// ---------------------------------------------------------------------------
// K4: partial_b[c] = sum_{i in block rows} w_i * value[i][c]
// Streams 512 MB of value through LDS with GLOBAL_LOAD_ASYNC_TO_LDS_B128.
// Each wave owns 4 x 2KB staging buffers (64 KB/block); 3-row-deep pipeline:
// issue row k+3, s_wait_asynccnt 8 proves row k landed (in-order completion),
// consume via ds_load_b128. Cross-wave column reduce reuses the LDS at the end.
// No atomics -> bitwise deterministic across graph replays.
// ---------------------------------------------------------------------------
#define ROWS_PER_WAVE 64

__global__ __launch_bounds__(256) void outsum_kernel(const float* __restrict__ value,
                                                     const float* __restrict__ weights,
                                                     float* __restrict__ partial) {
  __shared__ __align__(16) float ldsbuf[16384];   // 64 KB: 8 waves * 4 bufs * 512 floats
  const int lane  = threadIdx.x & 31;
  const int wid   = threadIdx.x >> 5;
  const int rbase = blockIdx.x * (NKEYS / NB_OUT) + wid * ROWS_PER_WAVE;
  const unsigned wbase_f = wid * 2048;            // float index of this wave's region

  float4 acc[4];
#pragma unroll
  for (int s = 0; s < 4; ++s) acc[s] = make_float4(0.f, 0.f, 0.f, 0.f);

  auto issue = [&](int k) {
    const float* gp = value + (size_t)(rbase + k) * KD + lane * 4;
    const unsigned lb = (unsigned)(uintptr_t)&ldsbuf[wbase_f + (unsigned)(k & 3) * 512 + lane * 4];
#pragma unroll
    for (int s = 0; s < 4; ++s) {
      asm volatile("global_load_async_to_lds_b128 %0, %1, off"
                   :: "v"(lb + s * 512u),
                      "v"((unsigned long long)(uintptr_t)(gp + s * 128))
                   : "memory");
    }
  };
  auto consume = [&](int k) {
    const float w = weights[rbase + k];
    const unsigned fb = wbase_f + (unsigned)(k & 3) * 512;
#pragma unroll
    for (int s = 0; s < 4; ++s) {
      const float4 v = *(const float4*)&ldsbuf[fb + s * 128 + lane * 4];
      acc[s].x += w * v.x; acc[s].y += w * v.y;
      acc[s].z += w * v.z; acc[s].w += w * v.w;
    }
  };

  issue(0); issue(1); issue(2);
  for (int k = 0; k < ROWS_PER_WAVE - 2; ++k) {
    asm volatile("s_wait_asynccnt 0x8" ::: "memory");   // rows k+1,k+2 may remain
    consume(k);
    if (k < ROWS_PER_WAVE - 3) issue(k + 3);
  }
  asm volatile("s_wait_asynccnt 0x4" ::: "memory");
  consume(ROWS_PER_WAVE - 2);
  asm volatile("s_wait_asynccnt 0x0" ::: "memory");
  consume(ROWS_PER_WAVE - 1);

  // cross-wave reduction: reuse staging LDS (all async ops retired above)
  __syncthreads();
#pragma unroll
  for (int s = 0; s < 4; ++s)
    *(float4*)&ldsbuf[wid * 512 + s * 128 + lane * 4] = acc[s];
  __syncthreads();
  const int tid = threadIdx.x;
  float t0 = 0.f, t1 = 0.f;
#pragma unroll
  for (int w = 0; w < 8; ++w) {
    t0 += ldsbuf[w * 512 + tid];
    t1 += ldsbuf[w * 512 + tid + 256];
  }
  partial[(size_t)blockIdx.x * KD + tid]       = t0;
  partial[(size_t)blockIdx.x * KD + tid + 256] = t1;
}

// ---------------------------------------------------------------------------
// K5: out[c] = (sum_b partial[b][c]) / gsum ; gsum re-reduced in-kernel (4 KB)
// ---------------------------------------------------------------------------
__global__ __launch_bounds__(256) void finalize_kernel(const float* __restrict__ partial,
                                                       const float* __restrict__ blocksum,
                                                       float* __restrict__ out) {
  __shared__ float s[256];
  float a = 0.f;
  for (int i = threadIdx.x; i < NB_EXP; i += 256) a += blocksum[i];
  s[threadIdx.x] = a;
  __syncthreads();
  for (int off = 128; off >= 1; off >>= 1) {
    if (threadIdx.x < off) s[threadIdx.x] += s[threadIdx.x + off];
    __syncthreads();
  }
  const float inv = 1.0f / s[0];
  const int c = blockIdx.x * 256 + threadIdx.x;   // grid = 2 blocks -> 512 cols
  float t = 0.f;
  for (int b = 0; b < NB_OUT; ++b) t += partial[(size_t)b * KD + c];
  out[c] = t * inv;
}

// ---------------------------------------------------------------------------
extern "C" void kernel_launch(void* const* d_in, const int* in_sizes, int n_in,
                              void* d_out, int out_size, void* d_ws, size_t ws_size,
                              hipStream_t stream) {
  const float* query = (const float*)d_in[0];   // [256]
  const float* key   = (const float*)d_in[1];   // [N,512]
  const float* value = (const float*)d_in[2];   // [N,512]
  const float* W     = (const float*)d_in[3];   // [512,256]
  const float* bias  = (const float*)d_in[4];   // [512]
  float* out = (float*)d_out;                   // [512]

  float* ws = (float*)d_ws;
  float* q        = ws;                         // 512
  float* scores   = q + KD;                     // N (rewritten as weights each call)
  float* blockmax = scores + NKEYS;             // NB_SCORE
  float* blocksum = blockmax + NB_SCORE;        // NB_EXP
  float* partial  = blocksum + NB_EXP;          // NB_OUT * 512

  qproj_wmma<<<1, 256, 0, stream>>>(W, query, bias, q);
  score_kernel<<<NB_SCORE, 256, 0, stream>>>(key, q, scores, blockmax);
  expsum_kernel<<<NB_EXP, 256, 0, stream>>>(scores, blockmax, blocksum);
  outsum_kernel<<<NB_OUT, 256, 0, stream>>>(value, scores, partial);
  finalize_kernel<<<2, 256, 0, stream>>>(partial, blocksum, out);
}